// PoolingLayer_86277303042222
// MI455X (gfx1250) — compile-verified
//
#include <hip/hip_runtime.h>
#include <cfloat>
#include <cstdint>
#include <cstddef>

// ---------------------------------------------------------------------------
// Gather + max-pool: out[p, :] = max_k feat[nbr[p,k], :]
//   N = 100000, NP = 50000, K = 32, F = 128 (float32)
//
// Memory-bound (L2 gather): feature table (51.2 MB) fits in the 192 MB L2,
// gather reads ~819 MB from L2, HBM sees ~90 MB total (~4 us at 23.3 TB/s).
// Strategy: one wave32 per output point; each neighbor row = one wave-wide
// coalesced global_load_b128 (512 B); indices scalarized to SGPRs with
// v_readlane so loads use the saddr form; 4 accumulators for MLP;
// non-temporal index loads + non-temporal b128 output stores.
// WMMA does not apply (no matmul structure in a gather/max).
// ---------------------------------------------------------------------------

typedef float v4f __attribute__((ext_vector_type(4)));
typedef int   v4i __attribute__((vector_size(16)));   // matches async-LDS builtin param

#define KNB   32
#define FDIM  128
#define BLOCK 256   // 8 wave32s per workgroup

static __device__ __forceinline__ v4f vmax4(v4f a, v4f b) {
    v4f r;
    r.x = fmaxf(a.x, b.x);
    r.y = fmaxf(a.y, b.y);
    r.z = fmaxf(a.z, b.z);
    r.w = fmaxf(a.w, b.w);
    return r;
}

__global__ __launch_bounds__(BLOCK)
void pool_max_k32_f128(const float* __restrict__ feat,
                       const int*   __restrict__ nbr,
                       float*       __restrict__ out,
                       int np)
{
    const int tid  = blockIdx.x * BLOCK + threadIdx.x;
    const int p    = tid >> 5;            // one wave per output point
    const int lane = threadIdx.x & 31;    // lane -> float4 chunk of the row
    if (p >= np) return;                  // whole wave exits together

    // Each lane loads one of this wave's 32 neighbor indices (coalesced,
    // read-once -> non-temporal).
    const int myidx = __builtin_nontemporal_load(nbr + (size_t)p * KNB + lane);

    const v4f* __restrict__ fp = (const v4f*)feat;   // row = 32 x v4f

    v4f a0 = {-FLT_MAX, -FLT_MAX, -FLT_MAX, -FLT_MAX};
    v4f a1 = a0, a2 = a0, a3 = a0;

#pragma unroll
    for (int k = 0; k < KNB; k += 4) {
        // Scalarize the (wave-uniform) row indices into SGPRs.
        const unsigned j0 = (unsigned)__builtin_amdgcn_readlane(myidx, k + 0);
        const unsigned j1 = (unsigned)__builtin_amdgcn_readlane(myidx, k + 1);
        const unsigned j2 = (unsigned)__builtin_amdgcn_readlane(myidx, k + 2);
        const unsigned j3 = (unsigned)__builtin_amdgcn_readlane(myidx, k + 3);

        // 4 independent wave-wide 512B row reads (global_load_b128, saddr form).
        const v4f v0 = fp[(size_t)j0 * (FDIM / 4) + lane];
        const v4f v1 = fp[(size_t)j1 * (FDIM / 4) + lane];
        const v4f v2 = fp[(size_t)j2 * (FDIM / 4) + lane];
        const v4f v3 = fp[(size_t)j3 * (FDIM / 4) + lane];

        a0 = vmax4(a0, v0);
        a1 = vmax4(a1, v1);
        a2 = vmax4(a2, v2);
        a3 = vmax4(a3, v3);
    }

    const v4f r = vmax4(vmax4(a0, a1), vmax4(a2, a3));

    // Streamed write-once output: non-temporal b128 store.
    __builtin_nontemporal_store(r, (v4f*)out + (size_t)p * (FDIM / 4) + lane);
}

// Generic fallback for unexpected shapes (one thread per output element).
__global__ __launch_bounds__(BLOCK)
void pool_max_generic(const float* __restrict__ feat,
                      const int*   __restrict__ nbr,
                      float*       __restrict__ out,
                      int np, int K, int F)
{
    const int idx = blockIdx.x * BLOCK + threadIdx.x;
    if (idx >= np * F) return;
    const int p = idx / F;
    const int f = idx - p * F;
    float m = -FLT_MAX;
    for (int k = 0; k < K; ++k) {
        const int j = nbr[(size_t)p * K + k];
        m = fmaxf(m, feat[(size_t)j * F + f]);
    }
    out[idx] = m;
}

// ---------------------------------------------------------------------------
// Compile-probe for the gfx1250 async-to-LDS path (never launched).
// Round-1 diagnostic confirmed the builtin exists and takes v4i* pointers.
// If this lowers to global_load_async_to_lds_b128 / s_wait_asynccnt in the
// histogram, a double-buffered async-gather variant becomes an option.
// ---------------------------------------------------------------------------
#if defined(__has_builtin)
#if __has_builtin(__builtin_amdgcn_global_load_async_to_lds_b128)
#define HAVE_ASYNC_LDS 1
__global__ void cdna5_async_probe(const float* __restrict__ g,
                                  float* __restrict__ o)
{
    __shared__ v4f stage[64];
    __builtin_amdgcn_global_load_async_to_lds_b128(
        (v4i*)g + threadIdx.x, (v4i*)&stage[threadIdx.x & 63], 0, 0);
#if __has_builtin(__builtin_amdgcn_s_wait_asynccnt)
    __builtin_amdgcn_s_wait_asynccnt(0);
#endif
    __syncthreads();
    o[threadIdx.x] = stage[threadIdx.x & 63].x;
}
#endif
#endif

// ---------------------------------------------------------------------------

extern "C" void kernel_launch(void* const* d_in, const int* in_sizes, int n_in,
                              void* d_out, int out_size, void* d_ws, size_t ws_size,
                              hipStream_t stream)
{
    (void)d_ws; (void)ws_size; (void)n_in;

    // setup_inputs() order: points (N,3) f32 [unused], features (N,F) f32,
    // neighbor_indices (NP,K) int32.
    const float* feat = (const float*)d_in[1];
    const int*   nbr  = (const int*)d_in[2];
    float*       out  = (float*)d_out;

    const int N  = in_sizes[0] / 3;
    const int F  = (N > 0) ? in_sizes[1] / N : FDIM;
    const int NP = (F > 0) ? out_size / F : 0;
    const int K  = (NP > 0) ? in_sizes[2] / NP : KNB;

    if (NP <= 0) return;

    if (F == FDIM && K == KNB) {
        const int total  = NP * 32;                    // one wave per point
        const int blocks = (total + BLOCK - 1) / BLOCK;
        pool_max_k32_f128<<<blocks, BLOCK, 0, stream>>>(feat, nbr, out, NP);
    } else {
        const int total  = NP * F;
        const int blocks = (total + BLOCK - 1) / BLOCK;
        pool_max_generic<<<blocks, BLOCK, 0, stream>>>(feat, nbr, out, NP, K, F);
    }
}